// DPTTransformerLiftSplatShoot_3169685865335
// MI455X (gfx1250) — compile-verified
//
#include <hip/hip_runtime.h>
#include <math.h>

// ---------------------------------------------------------------------------
// LSS BEV pooling for MI455X (gfx1250, wave32).
//   B=4 N=4 D=41 FH=16 FW=44 C=64 ; BEV grid 256x256x1
//   Np = 461,824 points, out = 4 x 64 x 256 x 256 f32 (67 MB, fits in 192MB L2)
// Design: 1 wave per 16 points. Geometry = two V_WMMA_F32_16X16X4_F32 affine
// transforms (homogeneous coords) with the (x*z, y*z, z) pointwise step done
// via a lane shuffle between them. Splat = coalesced float2 feature loads +
// hardware global_atomic_add_f32 into the L2-resident output.
// ---------------------------------------------------------------------------

typedef __attribute__((ext_vector_type(2))) float v2f;
typedef __attribute__((ext_vector_type(8))) float v8f;

#define NCAM   16        // B*N
#define PPC    28864     // D*FH*FW points per camera (multiple of 16)
#define WPC    1804      // PPC/16 waves per camera
#define NWAVES 28864     // Np/16 total waves
#define C_CH   64
#define PLANE  65536     // NY*NX
#define FWW    44
#define HWROW  704       // FH*FW

__device__ __forceinline__ void inv3(const float* m, float* o) {
    float a = m[0], b = m[1], c = m[2];
    float d = m[3], e = m[4], f = m[5];
    float g = m[6], h = m[7], i = m[8];
    float A =  (e * i - f * h);
    float Bc = -(d * i - f * g);
    float Cc =  (d * h - e * g);
    float det = a * A + b * Bc + c * Cc;
    float r = 1.0f / det;
    o[0] = A * r;               o[1] = -(b * i - c * h) * r; o[2] =  (b * f - c * e) * r;
    o[3] = Bc * r;              o[4] =  (a * i - c * g) * r; o[5] = -(a * f - c * d) * r;
    o[6] = Cc * r;              o[7] = -(a * h - b * g) * r; o[8] =  (a * e - b * d) * r;
}

__global__ __launch_bounds__(256) void lss_zero(float4* __restrict__ out, int n4) {
    int i = blockIdx.x * blockDim.x + threadIdx.x;
    if (i < n4) out[i] = make_float4(0.f, 0.f, 0.f, 0.f);
}

__global__ __launch_bounds__(256) void lss_fused(
        const float* __restrict__ feat,        // (B,N,D,FH,FW,C)
        const float* __restrict__ rots,        // (B,N,3,3)
        const float* __restrict__ trans,       // (B,N,3)
        const float* __restrict__ intrins,     // (B,N,3,3)
        const float* __restrict__ post_rots,   // (B,N,3,3)
        const float* __restrict__ post_trans,  // (B,N,3)
        float* __restrict__ out)               // (B, C, 256, 256)
{
    const int tid  = blockIdx.x * blockDim.x + threadIdx.x;
    const int wave = tid >> 5;
    const int lane = tid & 31;
    if (wave >= NWAVES) return;                 // wave-uniform; exec stays all-1s inside

    const int pointBase = wave << 4;            // 16 consecutive points
    const int cam       = wave / WPC;           // camera index, uniform per wave
    const int bIdx      = cam >> 2;             // N = 4

    // ---- per-camera matrices (wave-uniform, ~100 flops, negligible) ----
    float R[9], Km[9], P[9], iK[9], iP[9], M[9], t3[3], pT[3], t1[3];
    #pragma unroll
    for (int i = 0; i < 9; ++i) {
        R[i]  = rots[cam * 9 + i];
        Km[i] = intrins[cam * 9 + i];
        P[i]  = post_rots[cam * 9 + i];
    }
    #pragma unroll
    for (int i = 0; i < 3; ++i) {
        t3[i] = trans[cam * 3 + i];
        pT[i] = post_trans[cam * 3 + i];
    }
    inv3(Km, iK);
    inv3(P, iP);
    #pragma unroll
    for (int r = 0; r < 3; ++r) {               // M = rots @ inv(intrins)
        #pragma unroll
        for (int c = 0; c < 3; ++c)
            M[r * 3 + c] = R[r * 3 + 0] * iK[0 * 3 + c]
                         + R[r * 3 + 1] * iK[1 * 3 + c]
                         + R[r * 3 + 2] * iK[2 * 3 + c];
        t1[r] = -(iP[r * 3 + 0] * pT[0] + iP[r * 3 + 1] * pT[1] + iP[r * 3 + 2] * pT[2]);
    }

    // ---- frustum coordinates for this lane's point column (lane & 15) ----
    const int r15  = lane & 15;
    const int pic  = (pointBase - cam * PPC) + r15;    // point-in-camera
    const int dbin = pic / HWROW;
    const int rem  = pic - dbin * HWROW;
    const int hh   = rem / FWW;
    const int ww   = rem - hh * FWW;
    const float fx = (float)ww * (703.0f / 43.0f);     // linspace(0,703,44)
    const float fy = (float)hh * 17.0f;                // linspace(0,255,16)
    const float fz = 2.0f + (float)dbin;               // depth bins 2..42

    const bool lo = (lane < 16);

    // B1 (4x16, homogeneous point): lanes 0-15 hold {K0,K1}={fx,fy},
    //                               lanes 16-31 hold {K2,K3}={fz,1}
    v2f b1;
    b1.x = lo ? fx : fz;
    b1.y = lo ? fy : 1.0f;
    // A1 (16x4): rows 0..2 = [ inv(post_rots) | -inv(post_rots)@post_trans ]
    v2f a1;
    a1.x = (r15 < 3) ? (lo ? iP[r15 * 3 + 0] : iP[r15 * 3 + 2]) : 0.0f;
    a1.y = (r15 < 3) ? (lo ? iP[r15 * 3 + 1] : t1[r15])          : 0.0f;

    v8f zc = {};
    // p1 = inv(post_rots) @ (frustum - post_trans)
    v8f c1 = __builtin_amdgcn_wmma_f32_16x16x4_f32(false, a1, false, b1,
                                                   (short)0, zc, false, false);
    // lane l<16: c1[0]=p1x, c1[1]=p1y, c1[2]=p1z of point l

    // pointwise (x*z, y*z, z): rebuild B layout, shuffling z to lanes 16-31
    float zsh = __shfl(c1[2], r15, 32);
    v2f b2;
    b2.x = lo ? c1[0] * c1[2] : zsh;
    b2.y = lo ? c1[1] * c1[2] : 1.0f;
    // A2: rows 0..2 = [ rots@inv(intrins) | trans ]
    v2f a2;
    a2.x = (r15 < 3) ? (lo ? M[r15 * 3 + 0] : M[r15 * 3 + 2]) : 0.0f;
    a2.y = (r15 < 3) ? (lo ? M[r15 * 3 + 1] : t3[r15])         : 0.0f;

    // geom = M @ p2 + trans
    v8f c2 = __builtin_amdgcn_wmma_f32_16x16x4_f32(false, a2, false, b2,
                                                   (short)0, zc, false, false);

    // ---- voxelize (valid in lanes 0-15) ----
    const int vx = (int)floorf((c2[0] + 51.2f) / 0.4f);
    const int vy = (int)floorf((c2[1] + 51.2f) / 0.4f);
    const int vz = (int)floorf((c2[2] + 10.0f) / 20.0f);
    const int lin = (vx >= 0 && vx < 256 && vy >= 0 && vy < 256 && vz == 0)
                        ? ((vy << 8) + vx) : -1;

    // ---- splat: 16 points, 64 channels; 2 channels per lane ----
    float* outB = out + (size_t)bIdx * ((size_t)C_CH * PLANE);
    const float* fbase = feat + (size_t)pointBase * C_CH + (size_t)(lane << 1);
    const int c0 = lane << 1;
    for (int p = 0; p < 16; ++p) {
        const int lin_p = __shfl(lin, p, 32);   // wave-uniform skip
        if (lin_p < 0) continue;
        const float2 f = *(const float2*)(fbase + p * C_CH);  // 256B coalesced/wave
        unsafeAtomicAdd(outB + (size_t)c0 * PLANE + lin_p, f.x);
        unsafeAtomicAdd(outB + (size_t)(c0 + 1) * PLANE + lin_p, f.y);
    }
}

extern "C" void kernel_launch(void* const* d_in, const int* in_sizes, int n_in,
                              void* d_out, int out_size, void* d_ws, size_t ws_size,
                              hipStream_t stream) {
    const float* x          = (const float*)d_in[0];
    const float* rots       = (const float*)d_in[1];
    const float* trans      = (const float*)d_in[2];
    const float* intrins    = (const float*)d_in[3];
    const float* post_rots  = (const float*)d_in[4];
    const float* post_trans = (const float*)d_in[5];
    float* out = (float*)d_out;

    // zero the 67MB BEV output (out_size = 4*64*256*256, multiple of 4)
    const int n4 = out_size >> 2;
    hipLaunchKernelGGL(lss_zero, dim3((n4 + 255) / 256), dim3(256), 0, stream,
                       (float4*)out, n4);

    // 28,864 waves * 16 points = 461,824 points; 8 waves/block -> 3,608 blocks
    hipLaunchKernelGGL(lss_fused, dim3(NWAVES / 8), dim3(256), 0, stream,
                       x, rots, trans, intrins, post_rots, post_trans, out);
}